// SkipHashingMemory_34763465294229
// MI455X (gfx1250) — compile-verified
//
#include <hip/hip_runtime.h>

typedef __attribute__((ext_vector_type(2))) float v2f;
typedef __attribute__((ext_vector_type(8))) float v8f;

#define WMMA_F32(a, b, c) \
    __builtin_amdgcn_wmma_f32_16x16x4_f32(false, (a), false, (b), (short)0, (c), false, false)

#if defined(__AMDGCN__) && __has_builtin(__builtin_amdgcn_tensor_load_to_lds) && \
    __has_builtin(__builtin_amdgcn_s_wait_tensorcnt)
#define USE_TDM 1
#else
#define USE_TDM 0
#endif

#if USE_TDM
typedef __attribute__((ext_vector_type(4))) unsigned int u32x4;
typedef __attribute__((ext_vector_type(4))) int i32x4;
typedef __attribute__((ext_vector_type(8))) int i32x8;

typedef __attribute__((address_space(3))) const void lds_cvoid;
#define LDS_OFF(p) ((unsigned)(unsigned long long)(lds_cvoid*)(p))

// Build a 2-D TDM descriptor (D#) and issue TENSOR_LOAD_TO_LDS.
// Tile: tile_d0 x tile_d1 f32 elements, row stride `stride0` elements.
// LDS pad: +4 DWORDs every 16 DWORDs -> padded [row][20] layout for tile_d0=16.
__device__ __forceinline__ void tdm_load_2d(unsigned lds_off, unsigned long long gaddr,
                                            unsigned tensor_d0, unsigned tensor_d1,
                                            unsigned stride0, unsigned tile_d0,
                                            unsigned tile_d1, int pad_en) {
    u32x4 g0;
    g0.x = 1u;                                     // count=1, is_restore=0, gather off
    g0.y = lds_off;                                // lds_addr (bytes)
    g0.z = (unsigned)(gaddr & 0xffffffffull);      // global_addr[31:0]
    g0.w = (unsigned)((gaddr >> 32) & 0x1ffffffull) | (2u << 30);  // [56:32] | type=2
    i32x8 g1;
    unsigned w0 = (2u << 16);                      // data_size = 4 bytes
    if (pad_en) w0 |= (1u << 20) | (3u << 22) | (3u << 25);  // pad: every 16 DW, +4 DW
    g1[0] = (int)w0;
    g1[1] = (int)((tensor_d0 & 0xffffu) << 16);                                   // dim0 lo
    g1[2] = (int)(((tensor_d0 >> 16) & 0xffffu) | ((tensor_d1 & 0xffffu) << 16)); // dim0 hi|dim1 lo
    g1[3] = (int)(((tensor_d1 >> 16) & 0xffffu) | ((tile_d0 & 0xffffu) << 16));   // dim1 hi|tile0
    g1[4] = (int)(tile_d1 & 0xffffu);                                             // tile1, tile2=0
    g1[5] = (int)stride0;                                                         // stride0[31:0]
    g1[6] = 0;                                                                    // stride0 hi, stride1 lo
    g1[7] = 0;
    i32x4 gz4 = {0, 0, 0, 0};
    i32x8 gz8 = {0, 0, 0, 0, 0, 0, 0, 0};
    __builtin_amdgcn_tensor_load_to_lds(g0, g1, gz4, gz4, gz8, 0);
}
#endif

// ---------------------------------------------------------------------------
// Kernel 1: q = x @ w_q + b_q     (8192 x 1024 x 1024, fp32 WMMA)
// block = 256 thr (8 waves), tile 128x64, K-chunk 16
// A staged via TDM into padded [m][20] (b64 fragment loads, conflict-free)
// ---------------------------------------------------------------------------
__global__ __launch_bounds__(256) void gemm_q_kernel(const float* __restrict__ x,
                                                     const float* __restrict__ wq,
                                                     const float* __restrict__ bq,
                                                     float* __restrict__ q) {
    __shared__ float As[128][20];   // [m][k-pair contiguous], 20-float padded rows
    __shared__ float Bs[16][64];    // [k][n]
    const int tid  = threadIdx.x;
    const int lane = tid & 31;
    const int wave = tid >> 5;
    const int lm   = lane & 15;
    const int koff = (lane < 16) ? 0 : 2;
    const int m0 = blockIdx.x * 128;
    const int n0 = blockIdx.y * 64;

    v8f acc[4] = {};
    for (int kb = 0; kb < 1024; kb += 16) {
#if USE_TDM
        if (tid < 32) {
            unsigned long long ga = (unsigned long long)(uintptr_t)x +
                                    ((unsigned long long)m0 * 1024ull + (unsigned)kb) * 4ull;
            tdm_load_2d(LDS_OFF(&As[0][0]), ga, 16u, 128u, 1024u, 16u, 128u, 1);
        }
#else
        #pragma unroll
        for (int i = 0; i < 2; ++i) {
            int f = i * 256 + tid;
            int r = f >> 2, cq = f & 3;
            const float4 v = *(const float4*)(x + (size_t)(m0 + r) * 1024 + kb + cq * 4);
            *(float4*)&As[r][cq * 4] = v;
        }
#endif
        {
            int r = tid >> 4, cq = tid & 15;
            const float4 v = *(const float4*)(wq + (size_t)(kb + r) * 1024 + n0 + cq * 4);
            *(float4*)&Bs[r][cq * 4] = v;
            if (kb + 16 < 1024)
                __builtin_prefetch(wq + (size_t)(kb + 16 + r) * 1024 + n0 + cq * 4, 0, 1);
        }
#if USE_TDM
        if (tid < 32) __builtin_amdgcn_s_wait_tensorcnt((short)0);
#endif
        __syncthreads();
        #pragma unroll
        for (int ks = 0; ks < 4; ++ks) {
            const int k = ks * 4 + koff;
            v2f a = *(const v2f*)&As[wave * 16 + lm][k];
            #pragma unroll
            for (int j = 0; j < 4; ++j) {
                v2f b; b.x = Bs[k][j * 16 + lm]; b.y = Bs[k + 1][j * 16 + lm];
                acc[j] = WMMA_F32(a, b, acc[j]);
            }
        }
        __syncthreads();
    }
    const int rbase = m0 + wave * 16 + ((lane < 16) ? 0 : 8);
    #pragma unroll
    for (int j = 0; j < 4; ++j) {
        const int col = n0 + j * 16 + lm;
        const float bias = bq[col];
        #pragma unroll
        for (int r = 0; r < 8; ++r)
            q[(size_t)(rbase + r) * 1024 + col] = acc[j][r] + bias;
    }
}

// ---------------------------------------------------------------------------
// Kernel 2: scores = q_slice @ keys[h][c]^T  (16 tok x 1024 keys x 128),
// then per-token top-32 (descending, stable) -> topS/topI
// keys chunk staged via TDM into padded [key][20]; scores overlay same LDS
// grid: (512 token-tiles, 8 hc)
// ---------------------------------------------------------------------------
#define KP 20
__global__ __launch_bounds__(256) void scores_topk_kernel(const float* __restrict__ q,
                                                          const float* __restrict__ keys,
                                                          float* __restrict__ topS,
                                                          int* __restrict__ topI) {
    __shared__ float Ks[1024 * KP];   // keys chunk [key][20] during GEMM; scores [tok][1024] after
    __shared__ float Qs[16][KP];      // [token][d-pair contiguous]
    __shared__ float TW[16][32];
    __shared__ int   TI[16][32];

    const int tid  = threadIdx.x;
    const int lane = tid & 31;
    const int wave = tid >> 5;
    const int lm   = lane & 15;
    const int koff = (lane < 16) ? 0 : 2;
    const int n0 = blockIdx.x * 16;
    const int hc = blockIdx.y;            // h*2 + c
    const int h = hc >> 1, c = hc & 1;
    const int qoff = h * 256 + c * 128;
    const float* kbase = keys + (size_t)hc * 1024 * 128;

    v8f acc[8] = {};
    for (int d0 = 0; d0 < 128; d0 += 16) {
        {   // Qs: 256 floats, 1 per thread
            int token = tid >> 4, dd = tid & 15;
            Qs[token][dd] = q[(size_t)(n0 + token) * 1024 + qoff + d0 + dd];
        }
#if USE_TDM
        if (tid < 32) {
            unsigned long long ga = (unsigned long long)(uintptr_t)kbase + (unsigned)d0 * 4ull;
            tdm_load_2d(LDS_OFF(&Ks[0]), ga, 16u, 1024u, 128u, 16u, 1024u, 1);
        }
#else
        #pragma unroll
        for (int i = 0; i < 16; ++i) {
            int f = i * 256 + tid;
            int key = f >> 2, dq = f & 3;
            const float4 v = *(const float4*)(kbase + (size_t)key * 128 + d0 + dq * 4);
            *(float4*)&Ks[key * KP + dq * 4] = v;
        }
#endif
#if USE_TDM
        if (tid < 32) __builtin_amdgcn_s_wait_tensorcnt((short)0);
#endif
        __syncthreads();
        #pragma unroll
        for (int ks = 0; ks < 4; ++ks) {
            const int k = ks * 4 + koff;
            v2f a = *(const v2f*)&Qs[lm][k];
            #pragma unroll
            for (int j = 0; j < 8; ++j) {
                const int nn = wave * 128 + j * 16 + lm;
                v2f b = *(const v2f*)&Ks[nn * KP + k];
                acc[j] = WMMA_F32(a, b, acc[j]);
            }
        }
        __syncthreads();
    }
    // dump scores into LDS (overlay key-staging buffer): Sc[token][key]
    const int trow = (lane < 16) ? 0 : 8;
    #pragma unroll
    for (int j = 0; j < 8; ++j)
        #pragma unroll
        for (int r = 0; r < 8; ++r)
            Ks[(trow + r) * 1024 + wave * 128 + j * 16 + lm] = acc[j][r];
    __syncthreads();
    // top-32 per token via sorted insertion (stable: ties keep lower index first)
    if (tid < 16) {
        const int token = tid;
        float* tw = TW[token];
        int*   ti = TI[token];
        for (int j = 0; j < 32; ++j) { tw[j] = -3.4e38f; ti[j] = 0; }
        for (int j = 0; j < 1024; ++j) {
            float s = Ks[token * 1024 + j];
            if (s > tw[31]) {
                int p = 31;
                while (p > 0 && tw[p - 1] < s) { tw[p] = tw[p - 1]; ti[p] = ti[p - 1]; --p; }
                tw[p] = s; ti[p] = j;
            }
        }
    }
    __syncthreads();
    #pragma unroll
    for (int i = 0; i < 2; ++i) {
        int f = i * 256 + tid;
        int token = f >> 5, kk = f & 31;
        size_t o = (((size_t)(n0 + token) * 4 + h) * 2 + c) * 32 + kk;
        topS[o] = TW[token][kk];
        topI[o] = TI[token][kk];
    }
}

// ---------------------------------------------------------------------------
// Kernel 3: per-token: merge top-32 of 32x32 Cartesian sums, softmax,
// gather values rows, weighted sum, accumulate over 4 heads -> mbuf[N][32]
// one wave per token; block = 8 waves
// ---------------------------------------------------------------------------
__global__ __launch_bounds__(256) void combine_kernel(const float* __restrict__ topS,
                                                      const int* __restrict__ topI,
                                                      const float* __restrict__ values,
                                                      float* __restrict__ mbuf) {
    __shared__ float s2s[8][32];
    __shared__ int   i2s[8][32];
    __shared__ float vrows[8][32][33];
    const int tid = threadIdx.x, lane = tid & 31, w = tid >> 5;
    const int n = blockIdx.x * 8 + w;
    float macc = 0.f;
    for (int h = 0; h < 4; ++h) {
        size_t base = (((size_t)n * 4 + h) * 2) * 32;
        float s1 = topS[base + lane];
        int   i1 = topI[base + lane];
        s2s[w][lane] = topS[base + 32 + lane];
        i2s[w][lane] = topI[base + 32 + lane];
        __syncthreads();
        // 32-way merge of 32 sorted candidate streams (s1[lane] + s2[*] desc)
        int ptr = 0;
        float cand = s1 + s2s[w][0];
        float my_s = 0.f; int my_i = 0;
        for (int t = 0; t < 32; ++t) {
            float v = cand; int who = lane;
            #pragma unroll
            for (int off = 16; off > 0; off >>= 1) {
                float ov = __shfl_xor(v, off, 32);
                int   ow = __shfl_xor(who, off, 32);
                if (ov > v || (ov == v && ow < who)) { v = ov; who = ow; }
            }
            int wi1  = __shfl(i1, who, 32);
            int wptr = __shfl(ptr, who, 32);
            int wini = wi1 * 1024 + i2s[w][wptr];
            if (lane == t) { my_s = v; my_i = wini; }
            if (lane == who) {
                ++ptr;
                cand = (ptr < 32) ? (s1 + s2s[w][ptr]) : -3.4e38f;
            }
        }
        // softmax (sorted desc -> lane 0 holds max)
        float mx = __shfl(my_s, 0, 32);
        float e = __expf(my_s - mx);
        float sum = e;
        #pragma unroll
        for (int off = 16; off > 0; off >>= 1) sum += __shfl_xor(sum, off, 32);
        float wgt = e / sum;
        // gather value row (L2-resident table), scale, transpose-reduce in LDS
        const float* vr = values + (size_t)my_i * 32;
        #pragma unroll
        for (int d = 0; d < 32; ++d) vrows[w][lane][d] = wgt * vr[d];
        __syncthreads();
        float col = 0.f;
        #pragma unroll
        for (int l = 0; l < 32; ++l) col += vrows[w][l][lane];
        macc += col;
        __syncthreads();
    }
    mbuf[(size_t)n * 32 + lane] = macc;
}

// ---------------------------------------------------------------------------
// Kernel 4: out = x + mbuf @ w_reproj + b_reproj   (8192 x 1024 x 32, WMMA)
// padded [row][36] layouts -> single b64 fragment loads
// ---------------------------------------------------------------------------
__global__ __launch_bounds__(256) void reproj_kernel(const float* __restrict__ mbuf,
                                                     const float* __restrict__ wr,
                                                     const float* __restrict__ br,
                                                     const float* __restrict__ x,
                                                     float* __restrict__ out) {
    __shared__ float As[128][36];   // [m][k]
    __shared__ float Bs[64][36];    // [n][k]
    const int tid  = threadIdx.x;
    const int lane = tid & 31;
    const int wave = tid >> 5;
    const int lm   = lane & 15;
    const int koff = (lane < 16) ? 0 : 2;
    const int m0 = blockIdx.x * 128;
    const int n0 = blockIdx.y * 64;

    #pragma unroll
    for (int i = 0; i < 4; ++i) {   // 128x32 floats = 1024 float4
        int f = i * 256 + tid;
        int r = f >> 3, kq = f & 7;
        const float4 v = *(const float4*)(mbuf + (size_t)(m0 + r) * 32 + kq * 4);
        *(float4*)&As[r][kq * 4] = v;
    }
    #pragma unroll
    for (int i = 0; i < 8; ++i) {   // 32x64 floats, scatter into [n][k]
        int f = i * 256 + tid;
        int r = f >> 6, j = f & 63;
        Bs[j][r] = wr[(size_t)r * 1024 + n0 + j];
    }
    __syncthreads();

    v8f acc[4] = {};
    #pragma unroll
    for (int ks = 0; ks < 8; ++ks) {
        const int k = ks * 4 + koff;
        v2f a = *(const v2f*)&As[wave * 16 + lm][k];
        #pragma unroll
        for (int j = 0; j < 4; ++j) {
            v2f b = *(const v2f*)&Bs[j * 16 + lm][k];
            acc[j] = WMMA_F32(a, b, acc[j]);
        }
    }
    const int rbase = m0 + wave * 16 + ((lane < 16) ? 0 : 8);
    #pragma unroll
    for (int j = 0; j < 4; ++j) {
        const int col = n0 + j * 16 + lm;
        const float bias = br[col];
        #pragma unroll
        for (int r = 0; r < 8; ++r) {
            size_t o = (size_t)(rbase + r) * 1024 + col;
            out[o] = x[o] + acc[j][r] + bias;
        }
    }
}

// ---------------------------------------------------------------------------
extern "C" void kernel_launch(void* const* d_in, const int* in_sizes, int n_in,
                              void* d_out, int out_size, void* d_ws, size_t ws_size,
                              hipStream_t stream) {
    const float* x      = (const float*)d_in[0];
    const float* wq     = (const float*)d_in[1];
    const float* bq     = (const float*)d_in[2];
    const float* keys   = (const float*)d_in[3];
    const float* values = (const float*)d_in[4];
    const float* wr     = (const float*)d_in[5];
    const float* br     = (const float*)d_in[6];
    float* out = (float*)d_out;

    // workspace layout (floats): q[8192*1024] | topS[8192*4*2*32] | topI | mbuf[8192*32]
    float* q    = (float*)d_ws;
    float* topS = q + (size_t)8192 * 1024;
    int*   topI = (int*)(topS + (size_t)8192 * 4 * 2 * 32);
    float* mbuf = (float*)(topI + (size_t)8192 * 4 * 2 * 32);

    gemm_q_kernel<<<dim3(64, 16), 256, 0, stream>>>(x, wq, bq, q);
    scores_topk_kernel<<<dim3(512, 8), 256, 0, stream>>>(q, keys, topS, topI);
    combine_kernel<<<dim3(1024), 256, 0, stream>>>(topS, topI, values, mbuf);
    reproj_kernel<<<dim3(64, 16), 256, 0, stream>>>(mbuf, wr, br, x, out);
}